// Aggregator_44186623541853
// MI455X (gfx1250) — compile-verified
//
#include <hip/hip_runtime.h>

#define B_TOT 16384

typedef __attribute__((ext_vector_type(2))) float v2f;
typedef __attribute__((ext_vector_type(8))) float v8f;
typedef __attribute__((ext_vector_type(4))) unsigned int v4u;
typedef __attribute__((ext_vector_type(8))) int v8i_;
typedef __attribute__((ext_vector_type(4))) int v4i_;

#define RSTRIDE 68     // R rows in LDS: A-operand reads hit 16 distinct banks (4m residues)
#define WB_STRIDE 160  // W1_bot k-pair-interleaved rows: half-waves 32 banks apart

#if __has_builtin(__builtin_amdgcn_tensor_load_to_lds) && __has_builtin(__builtin_amdgcn_s_wait_tensorcnt)
#define HAVE_TDM 1
#else
#define HAVE_TDM 0
#endif

#if HAVE_TDM
// Issue a TDM load of one contiguous 4096-dword tile -> LDS, inserting 4 pad
// dwords every 64 dwords (re-creating the 68-float padded row stride).
__device__ __forceinline__ void tdm_load_R(const float* gptr, unsigned lds_byte) {
    unsigned long long ga = (unsigned long long)(uintptr_t)gptr;
    v4u g0;
    g0.x = 1u;                                       // count=1 (valid descriptor)
    g0.y = lds_byte;                                 // lds_addr
    g0.z = (unsigned)(ga & 0xFFFFFFFFu);             // global_addr[31:0]
    g0.w = (unsigned)((ga >> 32) & 0x1FFFFFFu)       // global_addr[56:32]
         | (2u << 30);                               // type=2 ("image")
    v8i_ g1;
    g1[0] = (int)((2u << 16)                         // data_size=4B
                | (1u << 20)                         // pad_enable
                | (5u << 22)                         // pad_interval: every 64 dwords
                | (3u << 25));                       // pad_amount: 4 dwords
    g1[1] = (int)(0x1000u << 16);                    // tensor_dim0 = 4096 (low 16)
    g1[2] = (int)(1u << 16);                         // tensor_dim1 = 1
    g1[3] = (int)(4096u << 16);                      // tile_dim0 = 4096
    g1[4] = 1;                                       // tile_dim1 = 1
    g1[5] = 4096;                                    // tensor_dim0_stride = 4096
    g1[6] = 0;
    g1[7] = 0;
    v4i_ z4 = {0, 0, 0, 0};
#if defined(__clang_major__) && __clang_major__ >= 23
    v8i_ z8 = {0, 0, 0, 0, 0, 0, 0, 0};
    __builtin_amdgcn_tensor_load_to_lds(g0, g1, z4, z4, z8, 0);
#else
    __builtin_amdgcn_tensor_load_to_lds(g0, g1, z4, z4, 0);
#endif
}
#endif

__global__ __launch_bounds__(128) void aggregator_kernel(
    const float* __restrict__ selfv,     // [B,1,64]
    const float* __restrict__ neigh_v,   // [B,1,64,64]
    const float* __restrict__ neigh_r,   // [B,1,64,64]
    const float* __restrict__ user_emb,  // [B,64]
    const float* __restrict__ W1,        // [128,64]
    const float* __restrict__ b1,        // [64]
    const float* __restrict__ w2,        // [64]
    const float* __restrict__ b2,        // [1]
    float* __restrict__ out)             // [B,1,128]
{
    __shared__ float sWb[32 * WB_STRIDE];   // W1_bot, k-pair interleaved
    __shared__ float sW1t[64 * 64];
    __shared__ float sR[2][64 * RSTRIDE];   // double-buffered R tile
    __shared__ float sUE[64];
    __shared__ float sU1[64];
    __shared__ float sW2[64];
    __shared__ float sS[64];
    __shared__ float sP[64];
    __shared__ float sPart[128];

    const int t    = threadIdx.x;
    const int lane = t & 31;      // wave32
    const int wv   = t >> 5;      // wave id 0..3
    const int l16  = lane & 15;
    const int hh   = lane >> 4;   // half-wave

    // ---- one-time staging (persistent block) ----
    for (int i = t; i < 4096; i += 128) sW1t[i] = W1[i];
    for (int i = t; i < 4096; i += 128) {
        int k = i >> 6, c = i & 63;
        // pair-interleave: (k, k+1) at adjacent floats -> single ds_load_b64 B operand
        sWb[(k >> 1) * WB_STRIDE + c * 2 + (k & 1)] = W1[4096 + i];
    }
    if (t < 64) sW2[t] = w2[t];

#if HAVE_TDM
    const unsigned ldsR0 = (unsigned)(uintptr_t)&sR[0][0];
    const unsigned ldsR1 = (unsigned)(uintptr_t)&sR[1][0];
    if (wv == 0) tdm_load_R(neigh_r + (size_t)blockIdx.x * 4096, ldsR0);  // prologue
#endif
    __syncthreads();

    const float b2v = b2[0];

    int parity = 0;
    for (int b = blockIdx.x; b < B_TOT; b += gridDim.x, parity ^= 1) {
        const float* Vg = neigh_v + (size_t)b * 4096;
        const int nb = b + gridDim.x;

        // prefetch V: needed only at the end of this iteration
        __builtin_prefetch(Vg + t * 32, 0, 0);

#if HAVE_TDM
        if (wv == 0) {
            if (nb < B_TOT) {
                // kick next batch's R into the other buffer, then wait for current
                tdm_load_R(neigh_r + (size_t)nb * 4096, parity ? ldsR0 : ldsR1);
                __builtin_amdgcn_s_wait_tensorcnt(1);   // in-order: oldest (current) done
            } else {
                __builtin_amdgcn_s_wait_tensorcnt(0);
            }
        }
#else
        {   // fallback: cooperative load R -> LDS (padded stride)
            const float* Rg = neigh_r + (size_t)b * 4096;
            #pragma unroll
            for (int i = 0; i < 8; ++i) {
                const int idx = (i * 128 + t) * 4;
                const int r = idx >> 6, c = idx & 63;
                const float4 v = *(const float4*)(Rg + idx);
                *(float4*)(&sR[parity][r * RSTRIDE + c]) = v;
            }
        }
#endif
        if (t < 64) sUE[t] = user_emb[(size_t)b * 64 + t];
        __syncthreads();

        // ---- u1[d] = b1[d] + ue @ W1_top (broadcast row of the concat) ----
        if (t < 64) {
            float u = b1[t];
            #pragma unroll 8
            for (int k = 0; k < 64; ++k)
                u = fmaf(sUE[k], sW1t[k * 64 + t], u);
            sU1[t] = u;
        }

        // ---- main GEMM: Hpre = R @ W1_bot via V_WMMA_F32_16X16X4_F32 ----
        v8f acc[4] = {};
        const float* Rl = &sR[parity][0];
        const int aBase = (wv * 16 + l16) * RSTRIDE + 2 * hh;
        #pragma unroll
        for (int kk = 0; kk < 16; ++kk) {
            const v2f a = *(const v2f*)(&Rl[aBase + 4 * kk]);        // A[m, 4kk+2h..+1]
            const int kp = 2 * kk + hh;                              // k-pair row
            #pragma unroll
            for (int nt = 0; nt < 4; ++nt) {
                const v2f bv = *(const v2f*)(&sWb[kp * WB_STRIDE + (nt * 16 + l16) * 2]);
                acc[nt] = __builtin_amdgcn_wmma_f32_16x16x4_f32(
                    false, a, false, bv, (short)0, acc[nt], false, false);
            }
        }
        __syncthreads();   // sU1 ready; sS safe to write

        // ---- epilogue: relu(Hpre + u1) . w2, straight from accumulator layout ----
        float w2c[4], u1c[4];
        #pragma unroll
        for (int nt = 0; nt < 4; ++nt) {
            w2c[nt] = sW2[nt * 16 + l16];
            u1c[nt] = sU1[nt * 16 + l16];
        }
        #pragma unroll
        for (int v = 0; v < 8; ++v) {     // row n = 16*wv + 8*hh + v
            float dot = 0.f;
            #pragma unroll
            for (int nt = 0; nt < 4; ++nt) {
                float hval = acc[nt][v] + u1c[nt];
                hval = fmaxf(hval, 0.f);
                dot  = fmaf(hval, w2c[nt], dot);
            }
            dot += __shfl_xor(dot, 8, 32);   // reduce across 16-lane half
            dot += __shfl_xor(dot, 4, 32);
            dot += __shfl_xor(dot, 2, 32);
            dot += __shfl_xor(dot, 1, 32);
            if (l16 == v) sS[wv * 16 + hh * 8 + v] = dot;
        }
        __syncthreads();

        // ---- sigmoid + softmax over 64 neighbors (wave 0) ----
        if (wv == 0) {
            const float x0 = sS[lane], x1 = sS[lane + 32];
            const float s0 = 1.f / (1.f + __expf(-(x0 + b2v)));
            const float s1 = 1.f / (1.f + __expf(-(x1 + b2v)));
            float mx = fmaxf(s0, s1);
            #pragma unroll
            for (int off = 16; off >= 1; off >>= 1)
                mx = fmaxf(mx, __shfl_xor(mx, off, 32));
            const float e0 = __expf(s0 - mx);
            const float e1 = __expf(s1 - mx);
            float sm = e0 + e1;
            #pragma unroll
            for (int off = 16; off >= 1; off >>= 1)
                sm += __shfl_xor(sm, off, 32);
            const float inv = 1.f / sm;
            sP[lane]      = e0 * inv;
            sP[lane + 32] = e1 * inv;
        }
        __syncthreads();

        // ---- agg[d] = sum_n p[n] * V[n,d]  (coalesced stream of V) ----
        {
            const int d = t & 63, g = t >> 6;
            float a = 0.f;
            #pragma unroll 8
            for (int n = g * 32; n < g * 32 + 32; ++n)
                a = fmaf(sP[n], Vg[n * 64 + d], a);
            sPart[t] = a;
        }
        __syncthreads();
        if (t < 64) {
            out[(size_t)b * 128 + 64 + t] = sPart[t] + sPart[t + 64];
            out[(size_t)b * 128 + t]      = selfv[(size_t)b * 64 + t];
        }
        // next iteration's first __syncthreads orders LDS reuse
    }
}

extern "C" void kernel_launch(void* const* d_in, const int* in_sizes, int n_in,
                              void* d_out, int out_size, void* d_ws, size_t ws_size,
                              hipStream_t stream) {
    (void)in_sizes; (void)n_in; (void)out_size; (void)d_ws; (void)ws_size;
    const float* selfv = (const float*)d_in[0];
    const float* nv    = (const float*)d_in[1];
    const float* nr    = (const float*)d_in[2];
    const float* ue    = (const float*)d_in[3];
    const float* W1    = (const float*)d_in[4];
    const float* b1    = (const float*)d_in[5];
    const float* w2    = (const float*)d_in[6];
    const float* b2    = (const float*)d_in[7];
    float* out = (float*)d_out;

    dim3 grid(2048), block(128);
    hipLaunchKernelGGL(aggregator_kernel, grid, block, 0, stream,
                       selfv, nv, nr, ue, W1, b1, w2, b2, out);
}